// LSTM_61229053772057
// MI455X (gfx1250) — compile-verified
//
#include <hip/hip_runtime.h>

typedef float v2f __attribute__((ext_vector_type(2)));
typedef float v8f __attribute__((ext_vector_type(8)));

#define B_ 256
#define T_ 512
#define I_ 64
#define H_ 128
#define G_ 512  // 4*H

// ---------------- Kernel 0: zero h0 and c (ws is poisoned 0xAA) ----------------
__global__ void lstm_init(float* __restrict__ h0, float* __restrict__ c, int n) {
    int i = blockIdx.x * blockDim.x + threadIdx.x;
    if (i < n) { h0[i] = 0.f; c[i] = 0.f; }
}

// ---------------- Kernel 1: xg[t,b,g] = x[b,t,:] . W_ih[g,:] + b_ih[g] + b_hh[g]
// GEMM: M = B*T (row m = b*T + t, x rows are contiguous length-64), K = I, N = G.
// One wave = 16 M-rows x 64 N-cols (4 accumulators share the A fragment).
__global__ __launch_bounds__(256) void lstm_xgemm(
        const float* __restrict__ x, const float* __restrict__ Wih,
        const float* __restrict__ bih, const float* __restrict__ bhh,
        float* __restrict__ xg) {
    const int lane = threadIdx.x & 31;
    const int wave = threadIdx.x >> 5;
    const int task = blockIdx.x * 8 + wave;   // 65536 wave-tasks
    const int m0   = (task >> 3) * 16;        // 8192 M-tiles
    const int n0   = (task & 7) * 64;         // 8 N-groups of 64 cols
    const int lrow = lane & 15;
    const int lhi  = lane >> 4;               // half-wave select
    const int koff = 2 * lhi;                 // K sub-offset inside 4-wide K step

    v8f acc[4];
#pragma unroll
    for (int q = 0; q < 4; ++q) acc[q] = (v8f)0.f;

    const float* aRow = x + (size_t)(m0 + lrow) * I_ + koff;   // A: lane holds M=lrow, K=koff..koff+1
#pragma unroll 4
    for (int kk = 0; kk < I_; kk += 4) {
        v2f aF = *(const v2f*)(aRow + kk);
#pragma unroll
        for (int q = 0; q < 4; ++q) {
            const int g = n0 + q * 16 + lrow;                  // B: lane holds N=lrow, K=koff..koff+1
            v2f bF = *(const v2f*)(Wih + (size_t)g * I_ + kk + koff);
            acc[q] = __builtin_amdgcn_wmma_f32_16x16x4_f32(
                false, aF, false, bF, (short)0, acc[q], false, false);
        }
    }

#pragma unroll
    for (int q = 0; q < 4; ++q) {
        const int g    = n0 + q * 16 + lrow;
        const float bs = bih[g] + bhh[g];
#pragma unroll
        for (int r = 0; r < 8; ++r) {
            const int m = m0 + r + 8 * lhi;                    // C/D: VGPR r -> M = r + 8*hi
            const int t = m & (T_ - 1);
            const int b = m >> 9;                              // m / T
            xg[(size_t)t * (B_ * G_) + (size_t)b * G_ + g] = acc[q][r] + bs;
        }
    }
}

// ---------------- Kernel 2: one LSTM time step ----------------
// gates[b,g] = xg_t[b,g] + h_in[b,:] . W_hh[g,:]   (M=256, N=512, K=128)
// One wave owns a 16x16 (b,j) state tile and computes all 4 gate tiles for it,
// so the cell update is purely in-register (accumulator layouts align).
__global__ __launch_bounds__(256) void lstm_step(
        const float* __restrict__ Whh, const float* __restrict__ xgt,
        const float* __restrict__ hin, float* __restrict__ hout,
        float* __restrict__ c, float* __restrict__ out, int t) {
    const int lane = threadIdx.x & 31;
    const int wave = threadIdx.x >> 5;
    const int task = blockIdx.x * 8 + wave;   // 128 wave-tasks
    const int b0   = (task >> 3) * 16;        // 16 b-tiles
    const int j0   = (task & 7) * 16;         // 8 j-tiles
    const int lrow = lane & 15;
    const int lhi  = lane >> 4;
    const int koff = 2 * lhi;

    // Accumulators initialized with the precomputed input projection (includes biases)
    v8f acc[4];
#pragma unroll
    for (int q = 0; q < 4; ++q) {
        const int g = q * H_ + j0 + lrow;
#pragma unroll
        for (int r = 0; r < 8; ++r) {
            const int b = b0 + r + 8 * lhi;
            acc[q][r] = xgt[(size_t)b * G_ + g];
        }
    }

    const float* aRow = hin + (size_t)(b0 + lrow) * H_ + koff;
#pragma unroll 4
    for (int kk = 0; kk < H_; kk += 4) {
        v2f aF = *(const v2f*)(aRow + kk);
#pragma unroll
        for (int q = 0; q < 4; ++q) {
            const int g = q * H_ + j0 + lrow;
            v2f bF = *(const v2f*)(Whh + (size_t)g * H_ + kk + koff);
            acc[q] = __builtin_amdgcn_wmma_f32_16x16x4_f32(
                false, aF, false, bF, (short)0, acc[q], false, false);
        }
    }

    // Cell update: acc0=i, acc1=f, acc2=g, acc3=o (torch gate order)
    const int j = j0 + lrow;
#pragma unroll
    for (int r = 0; r < 8; ++r) {
        const int b = b0 + r + 8 * lhi;
        const float ig = 1.f / (1.f + expf(-acc[0][r]));
        const float fg = 1.f / (1.f + expf(-acc[1][r]));
        const float gg = tanhf(acc[2][r]);
        const float og = 1.f / (1.f + expf(-acc[3][r]));
        const size_t sidx = (size_t)b * H_ + j;
        const float cn = fg * c[sidx] + ig * gg;
        const float hn = og * tanhf(cn);
        c[sidx]    = cn;
        hout[sidx] = hn;
        out[(size_t)b * (T_ * H_) + (size_t)t * H_ + j] = hn;  // out is [B,T,H]
    }
}

extern "C" void kernel_launch(void* const* d_in, const int* in_sizes, int n_in,
                              void* d_out, int out_size, void* d_ws, size_t ws_size,
                              hipStream_t stream) {
    const float* x   = (const float*)d_in[0];   // [B,T,I]
    const float* Wih = (const float*)d_in[1];   // [4H,I]
    const float* Whh = (const float*)d_in[2];   // [4H,H]
    const float* bih = (const float*)d_in[3];   // [4H]
    const float* bhh = (const float*)d_in[4];   // [4H]
    float* out = (float*)d_out;                 // [B,T,H]

    // Workspace layout: xg [T,B,4H] f32 (256 MiB), then h0, h1, c ([B,H] each)
    float* xg = (float*)d_ws;
    float* h0 = xg + (size_t)T_ * B_ * G_;
    float* h1 = h0 + B_ * H_;
    float* c  = h1 + B_ * H_;

    lstm_init<<<(B_ * H_ + 255) / 256, 256, 0, stream>>>(h0, c, B_ * H_);

    // 65536 wave-tasks / 8 waves per block
    lstm_xgemm<<<8192, 256, 0, stream>>>(x, Wih, bih, bhh, xg);

    // Sequential recurrence: stream order provides the inter-step dependency;
    // h is double-buffered so no intra-step read/write race on the K operand.
    for (int t = 0; t < T_; ++t) {
        const float* hi = (t & 1) ? h1 : h0;
        float*       ho = (t & 1) ? h0 : h1;
        lstm_step<<<16, 256, 0, stream>>>(Whh, xg + (size_t)t * (B_ * G_), hi, ho, c, out, t);
    }
}